// InferSpConvModule_9268539425513
// MI455X (gfx1250) — compile-verified
//
#include <hip/hip_runtime.h>

typedef __attribute__((ext_vector_type(2))) float v2f;
typedef __attribute__((ext_vector_type(8))) float v8f;

#define KVOL 27
#define C 64                    // C_IN == C_OUT == 64
#define WAVES_PER_BLOCK 8
#define ROWS_PER_WAVE 16
#define LDS_PAIR_STRIDE 160     // 128 payload + 32 pad floats -> halves land on disjoint bank sets

__device__ __forceinline__ void atomic_add_f32(float* p, float v) {
    unsafeAtomicAdd(p, v);      // lowers to global_atomic_add_f32 (no return)
}

__global__ __launch_bounds__(256) void zero_kernel(float* __restrict__ out, int n4) {
    int i = blockIdx.x * blockDim.x + threadIdx.x;
    if (i < n4) {
        ((float4*)out)[i] = make_float4(0.f, 0.f, 0.f, 0.f);
    }
}

__global__ __launch_bounds__(256) void spconv_wmma_kernel(
    const float* __restrict__ in_feats,   // [N_ACT, 64]
    const float* __restrict__ weight,     // [27, 64, 64]
    const int*   __restrict__ in_index,   // [27, P]
    const int*   __restrict__ out_index,  // [27, P]
    float*       __restrict__ out,        // [N_ACT, 64] accumulation target
    int P, int tiles_per_k)
{
    // Weight for this kernel offset, pre-swizzled into WMMA-B fragment order:
    // element W[kr][n] (kr = K row 0..63, n = col 0..63) stored at
    //   pair = (kr>>2)*2 + ((kr>>1)&1)   (which 16x16x4 step + which half-wave)
    //   wlds[pair*LDS_PAIR_STRIDE + n*2 + (kr&1)]
    // so one ds_load_b64 yields the {K, K+1} pair a lane needs.
    __shared__ __align__(16) float wlds[32 * LDS_PAIR_STRIDE];

    const int k   = blockIdx.y;
    const int tid = threadIdx.x;

    // ---- stage weight[k] (4096 floats) into LDS, coalesced global reads ----
    #pragma unroll
    for (int e = 0; e < 16; ++e) {
        int i  = e * 256 + tid;            // 0..4095
        float w = weight[k * 4096 + i];
        int kr = i >> 6;
        int n  = i & 63;
        int pair = (kr >> 2) * 2 + ((kr >> 1) & 1);
        wlds[pair * LDS_PAIR_STRIDE + n * 2 + (kr & 1)] = w;
    }
    __syncthreads();

    const int wave = tid >> 5;             // wave32
    const int lane = tid & 31;
    const int tile = blockIdx.x * WAVES_PER_BLOCK + wave;
    if (tile >= tiles_per_k) return;       // wave-uniform: EXEC stays all-ones below

    const int rbase = tile * ROWS_PER_WAVE;
    const long idxbase = (long)k * P + rbase;
    const int ln  = lane & 15;
    const int sel = lane >> 4;             // 0 = lanes 0-15, 1 = lanes 16-31
    const int klo = sel * 2;               // A/B half-wave K offset within each K=4 step

    // Gather: lane ln of each half owns input row `row`; A fragment for step kk is
    // the contiguous fp32 pair in_feats[row][4*kk + klo .. +1] -> one global_load_b64.
    const int row = in_index[idxbase + ln];
    const float* ap = in_feats + (long)row * C + klo;

    v8f acc0 = {}, acc1 = {}, acc2 = {}, acc3 = {};

    #pragma unroll
    for (int kk = 0; kk < 16; ++kk) {
        v2f a = *(const v2f*)(ap + kk * 4);
        const float* bp = &wlds[(kk * 2 + sel) * LDS_PAIR_STRIDE + ln * 2];
        v2f b0 = *(const v2f*)(bp + 0 * 32);
        v2f b1 = *(const v2f*)(bp + 1 * 32);
        v2f b2 = *(const v2f*)(bp + 2 * 32);
        v2f b3 = *(const v2f*)(bp + 3 * 32);
        acc0 = __builtin_amdgcn_wmma_f32_16x16x4_f32(false, a, false, b0, (short)0, acc0, false, false);
        acc1 = __builtin_amdgcn_wmma_f32_16x16x4_f32(false, a, false, b1, (short)0, acc1, false, false);
        acc2 = __builtin_amdgcn_wmma_f32_16x16x4_f32(false, a, false, b2, (short)0, acc2, false, false);
        acc3 = __builtin_amdgcn_wmma_f32_16x16x4_f32(false, a, false, b3, (short)0, acc3, false, false);
    }

    // ---- scatter-add: C/D layout -> VGPR r holds M = r + 8*sel, N = ln + 16*nt ----
    const int hi = sel << 3;
    #pragma unroll
    for (int r = 0; r < 8; ++r) {
        int orow = out_index[idxbase + r + hi];
        float* op = out + (long)orow * C + ln;
        atomic_add_f32(op + 0,  acc0[r]);
        atomic_add_f32(op + 16, acc1[r]);
        atomic_add_f32(op + 32, acc2[r]);
        atomic_add_f32(op + 48, acc3[r]);
    }
}

__global__ __launch_bounds__(256) void bias_relu_kernel(
    float* __restrict__ out, const float* __restrict__ bias, int n4)
{
    int i = blockIdx.x * blockDim.x + threadIdx.x;
    if (i >= n4) return;
    float4 v = ((float4*)out)[i];
    int c = (i * 4) & 63;
    v.x = fmaxf(v.x + bias[c + 0], 0.f);
    v.y = fmaxf(v.y + bias[c + 1], 0.f);
    v.z = fmaxf(v.z + bias[c + 2], 0.f);
    v.w = fmaxf(v.w + bias[c + 3], 0.f);
    ((float4*)out)[i] = v;
}

extern "C" void kernel_launch(void* const* d_in, const int* in_sizes, int n_in,
                              void* d_out, int out_size, void* d_ws, size_t ws_size,
                              hipStream_t stream) {
    const float* in_feats  = (const float*)d_in[0];
    const float* weight    = (const float*)d_in[1];
    const float* bias      = (const float*)d_in[2];
    const int*   in_index  = (const int*)d_in[3];
    const int*   out_index = (const int*)d_in[4];
    float* out = (float*)d_out;

    const int P = in_sizes[3] / KVOL;              // 60000
    const int tiles_per_k = P / ROWS_PER_WAVE;     // 3750 (P divisible by 16)
    const int n4 = out_size / 4;                   // 3.2M float4s

    // 1) zero the accumulation target (d_out is poisoned by the harness)
    zero_kernel<<<(n4 + 255) / 256, 256, 0, stream>>>(out, n4);

    // 2) gather + WMMA GEMM + scatter-add
    dim3 grid((tiles_per_k + WAVES_PER_BLOCK - 1) / WAVES_PER_BLOCK, KVOL);
    spconv_wmma_kernel<<<grid, 256, 0, stream>>>(in_feats, weight, in_index, out_index,
                                                 out, P, tiles_per_k);

    // 3) fused bias + ReLU in place
    bias_relu_kernel<<<(n4 + 255) / 256, 256, 0, stream>>>(out, bias, n4);
}